// HeteroGNNs_15247133900891
// MI455X (gfx1250) — compile-verified
//
#include <hip/hip_runtime.h>
#include <hip/hip_bf16.h>

typedef __attribute__((ext_vector_type(2))) float v2f;
typedef __attribute__((ext_vector_type(8))) float v8f;

#define F_DIM 128
#define C_DIM 64

// ---------------------------------------------------------------------------
// Dense transform: H[N,64] = X[N,K] @ W[K,64] using V_WMMA_F32_16X16X4_F32.
// One wave32 computes a 16-row x 64-col strip (4 accumulator tiles of v8f).
// W staged in LDS in PAIR-INTERLEAVED layout so each B fragment
// {W[ka][col], W[ka+1][col]} is a single aligned ds_load_b64:
//   sW[(k>>1)*128 + col*2 + (k&1)] = W[k][col]
// A 16x4 f32 layout: lanes 0-15 hold K={kk,kk+1}, lanes 16-31 hold K={kk+2,kk+3}.
// B 4x16 layout mirrors the K split across lane halves.
// C/D: VGPR r holds M=r (lanes 0-15) and M=r+8 (lanes 16-31), N=lane&15.
// ---------------------------------------------------------------------------
__global__ void gemm_wmma_f32(const float* __restrict__ X,
                              const float* __restrict__ W,
                              float* __restrict__ H, int N, int K) {
    __shared__ float sW[F_DIM * C_DIM];   // up to 32 KB, pair-interleaved
    const int tid = threadIdx.x;
    for (int i = tid; i < K * C_DIM; i += 256) {
        const int k = i >> 6, col = i & 63;
        sW[(k >> 1) * 128 + col * 2 + (k & 1)] = W[i];
    }
    __syncthreads();

    const int wave = tid >> 5;
    const int lane = tid & 31;
    const long row0 = ((long)blockIdx.x * 8 + wave) * 16;
    if (row0 >= N) return;                 // whole-wave uniform: EXEC stays all-ones

    const int m  = lane & 15;              // row within tile (A) / col within tile (B,D)
    const int kp = lane >> 4;              // K-pair selector
    const float* xrow = X + (row0 + m) * (long)K;

    v8f c0 = {}, c1 = {}, c2 = {}, c3 = {};
    for (int kk = 0; kk < K; kk += 4) {
        const int ka = kk + kp * 2;        // even
        v2f a = *(const v2f*)(xrow + ka);  // global_load_b64
        const v2f* bp = (const v2f*)(sW + (ka >> 1) * 128);
        v2f b0 = bp[m];                    // ds_load_b64 each
        v2f b1 = bp[16 + m];
        v2f b2 = bp[32 + m];
        v2f b3 = bp[48 + m];
        c0 = __builtin_amdgcn_wmma_f32_16x16x4_f32(false, a, false, b0, (short)0, c0, false, false);
        c1 = __builtin_amdgcn_wmma_f32_16x16x4_f32(false, a, false, b1, (short)0, c1, false, false);
        c2 = __builtin_amdgcn_wmma_f32_16x16x4_f32(false, a, false, b2, (short)0, c2, false, false);
        c3 = __builtin_amdgcn_wmma_f32_16x16x4_f32(false, a, false, b3, (short)0, c3, false, false);
    }

    // Store D tiles: element (M = r + 8*kp, N = m) lives in accumulator VGPR r.
    #pragma unroll
    for (int r = 0; r < 8; ++r) {
        const long orow = (row0 + r + 8 * kp) * C_DIM;
        H[orow +  0 + m] = c0[r];
        H[orow + 16 + m] = c1[r];
        H[orow + 32 + m] = c2[r];
        H[orow + 48 + m] = c3[r];
    }
}

// v[f] = sum_c W[f*C + c] * a[c]   (tiny: one block)
__global__ void matvec_w(const float* __restrict__ W, const float* __restrict__ a,
                         float* __restrict__ v, int Fdim) {
    int f = threadIdx.x;
    if (f >= Fdim) return;
    float s = 0.f;
    #pragma unroll
    for (int c = 0; c < C_DIM; ++c) s += W[f * C_DIM + c] * a[c];
    v[f] = s;
}

// out[n] = dot(X[n,:], v)  — one wave per node, shuffle reduction (wave32)
__global__ void rowdot(const float* __restrict__ X, const float* __restrict__ v,
                       float* __restrict__ out, int N, int K) {
    int g = blockIdx.x * blockDim.x + threadIdx.x;
    int n = g >> 5, lane = g & 31;
    if (n >= N) return;
    float s = 0.f;
    const float* xr = X + (long)n * K;
    for (int k = lane; k < K; k += 32) s += xr[k] * v[k];
    #pragma unroll
    for (int off = 16; off > 0; off >>= 1) s += __shfl_down(s, off, 32);
    if (lane == 0) out[n] = s;
}

__global__ void fill_f32(float* __restrict__ p, float val, int n) {
    int i = blockIdx.x * blockDim.x + threadIdx.x;
    if (i < n) p[i] = val;
}

// pass1: logits + segment-max via int-bit-trick float atomic max
__global__ void edge_pass1(const int* __restrict__ src, const int* __restrict__ dst,
                           const float* __restrict__ as, const float* __restrict__ ad,
                           float* __restrict__ logits, float* __restrict__ mbuf, int E) {
    int e = blockIdx.x * blockDim.x + threadIdx.x;
    if (e >= E) return;
    int s = src[e], d = dst[e];
    float l = as[s] + ad[d];
    l = l > 0.f ? l : 0.2f * l;            // leaky_relu(0.2)
    logits[e] = l;
    if (l >= 0.f) atomicMax((int*)(mbuf + d), __float_as_int(l));
    else          atomicMin((unsigned int*)(mbuf + d), __float_as_uint(l));
}

// pass2: exp(logit - max) in place + segment-sum of denominators
__global__ void edge_pass2(const int* __restrict__ dst, float* __restrict__ logits,
                           const float* __restrict__ mbuf, float* __restrict__ den, int E) {
    int e = blockIdx.x * blockDim.x + threadIdx.x;
    if (e >= E) return;
    int d = dst[e];
    float ex = __expf(logits[e] - mbuf[d]);
    logits[e] = ex;
    atomicAdd(den + d, ex);
}

// pass3: weighted scatter of source features. One wave per edge, 2 channels/lane.
__global__ void edge_pass3(const int* __restrict__ src, const int* __restrict__ dst,
                           const float* __restrict__ ex, const float* __restrict__ den,
                           const float* __restrict__ hs, float* __restrict__ acc, int E) {
    int g = blockIdx.x * blockDim.x + threadIdx.x;
    int e = g >> 5, lane = g & 31;
    if (e >= E) return;
    int s = src[e], d = dst[e];
    float w = ex[e] / (den[d] + 1e-16f);
    const float2 hv = ((const float2*)(hs + (long)s * C_DIM))[lane];
    float* arow = acc + (long)d * C_DIM + lane * 2;
    atomicAdd(arow,     w * hv.x);
    atomicAdd(arow + 1, w * hv.y);
}

// out = 0.5 * (acc + b1[c] + b2[c])
__global__ void combine(const float* __restrict__ acc, const float* __restrict__ b1,
                        const float* __restrict__ b2, float* __restrict__ out, int N) {
    int i = blockIdx.x * blockDim.x + threadIdx.x;
    if (i >= N * C_DIM) return;
    int c = i & (C_DIM - 1);
    out[i] = 0.5f * (acc[i] + b1[c] + b2[c]);
}

extern "C" void kernel_launch(void* const* d_in, const int* in_sizes, int n_in,
                              void* d_out, int out_size, void* d_ws, size_t ws_size,
                              hipStream_t stream) {
    const float* x_t      = (const float*)d_in[0];
    const float* x_d      = (const float*)d_in[1];
    const int*   ei_tt    = (const int*)d_in[2];
    const int*   ei_dt    = (const int*)d_in[3];
    const int*   ei_dd    = (const int*)d_in[4];
    const int*   ei_td    = (const int*)d_in[5];
    const float* W_tt     = (const float*)d_in[6];
    const float* a_tt_s   = (const float*)d_in[7];
    const float* a_tt_d   = (const float*)d_in[8];
    const float* b_tt     = (const float*)d_in[9];
    const float* W_dt_s   = (const float*)d_in[10];
    const float* W_dt_d   = (const float*)d_in[11];
    const float* a_dt_s   = (const float*)d_in[12];
    const float* a_dt_d   = (const float*)d_in[13];
    const float* b_dt     = (const float*)d_in[14];
    const float* W_dd     = (const float*)d_in[15];
    const float* a_dd_s   = (const float*)d_in[16];
    const float* a_dd_d   = (const float*)d_in[17];
    const float* b_dd     = (const float*)d_in[18];
    const float* W_td_s   = (const float*)d_in[19];
    const float* W_td_d   = (const float*)d_in[20];
    const float* a_td_s   = (const float*)d_in[21];
    const float* a_td_d   = (const float*)d_in[22];
    const float* b_td     = (const float*)d_in[23];

    const int NT = in_sizes[0] / F_DIM;
    const int ND = in_sizes[1] / F_DIM;
    const int E  = in_sizes[2] / 2;
    const int maxN = NT > ND ? NT : ND;

    // workspace layout (floats)
    float* ws    = (float*)d_ws;
    float* hs    = ws;                          // maxN*64
    float* acc   = hs   + (size_t)maxN * C_DIM; // maxN*64
    float* exb   = acc  + (size_t)maxN * C_DIM; // E
    float* mbuf  = exb  + (size_t)E;            // maxN
    float* den   = mbuf + (size_t)maxN;         // maxN
    float* as    = den  + (size_t)maxN;         // maxN
    float* ad    = as   + (size_t)maxN;         // maxN
    float* wvec  = ad   + (size_t)maxN;         // 128

    float* out_t = (float*)d_out;               // x_target_new [NT,64]
    float* out_d = out_t + (size_t)NT * C_DIM;  // x_drug_new   [ND,64]

    const int TB = 256;
    auto cdiv = [](long a, long b) { return (int)((a + b - 1) / b); };

    auto run_edges = [&](const int* ei, const float* as_, const float* ad_,
                         const float* hs_, float* acc_, int n_dst) {
        fill_f32<<<cdiv(n_dst, TB), TB, 0, stream>>>(mbuf, -1e30f, n_dst);
        fill_f32<<<cdiv(n_dst, TB), TB, 0, stream>>>(den, 0.f, n_dst);
        edge_pass1<<<cdiv(E, TB), TB, 0, stream>>>(ei, ei + E, as_, ad_, exb, mbuf, E);
        edge_pass2<<<cdiv(E, TB), TB, 0, stream>>>(ei + E, exb, mbuf, den, E);
        edge_pass3<<<cdiv((long)E * 32, TB), TB, 0, stream>>>(ei, ei + E, exb, den, hs_, acc_, E);
    };

    // ---------------- target update: tt + dt -> x_target_new ----------------
    fill_f32<<<cdiv((long)NT * C_DIM, TB), TB, 0, stream>>>(acc, 0.f, NT * C_DIM);

    // tt relation
    gemm_wmma_f32<<<cdiv(NT, 128), TB, 0, stream>>>(x_t, W_tt, hs, NT, F_DIM);
    rowdot<<<cdiv((long)NT * 32, TB), TB, 0, stream>>>(hs, a_tt_s, as, NT, C_DIM);
    matvec_w<<<1, F_DIM, 0, stream>>>(W_tt, a_tt_d, wvec, F_DIM);
    rowdot<<<cdiv((long)NT * 32, TB), TB, 0, stream>>>(x_t, wvec, ad, NT, F_DIM);
    run_edges(ei_tt, as, ad, hs, acc, NT);

    // dt relation (drug -> target)
    gemm_wmma_f32<<<cdiv(ND, 128), TB, 0, stream>>>(x_d, W_dt_s, hs, ND, F_DIM);
    rowdot<<<cdiv((long)ND * 32, TB), TB, 0, stream>>>(hs, a_dt_s, as, ND, C_DIM);
    matvec_w<<<1, F_DIM, 0, stream>>>(W_dt_d, a_dt_d, wvec, F_DIM);
    rowdot<<<cdiv((long)NT * 32, TB), TB, 0, stream>>>(x_t, wvec, ad, NT, F_DIM);
    run_edges(ei_dt, as, ad, hs, acc, NT);

    combine<<<cdiv((long)NT * C_DIM, TB), TB, 0, stream>>>(acc, b_tt, b_dt, out_t, NT);

    // ---------------- drug update: dd + td -> x_drug_new --------------------
    fill_f32<<<cdiv((long)ND * C_DIM, TB), TB, 0, stream>>>(acc, 0.f, ND * C_DIM);

    // dd relation
    gemm_wmma_f32<<<cdiv(ND, 128), TB, 0, stream>>>(x_d, W_dd, hs, ND, F_DIM);
    rowdot<<<cdiv((long)ND * 32, TB), TB, 0, stream>>>(hs, a_dd_s, as, ND, C_DIM);
    matvec_w<<<1, F_DIM, 0, stream>>>(W_dd, a_dd_d, wvec, F_DIM);
    rowdot<<<cdiv((long)ND * 32, TB), TB, 0, stream>>>(x_d, wvec, ad, ND, F_DIM);
    run_edges(ei_dd, as, ad, hs, acc, ND);

    // td relation (target_new -> drug); src features are C-dim (K=64)
    gemm_wmma_f32<<<cdiv(NT, 128), TB, 0, stream>>>(out_t, W_td_s, hs, NT, C_DIM);
    rowdot<<<cdiv((long)NT * 32, TB), TB, 0, stream>>>(hs, a_td_s, as, NT, C_DIM);
    matvec_w<<<1, F_DIM, 0, stream>>>(W_td_d, a_td_d, wvec, F_DIM);
    rowdot<<<cdiv((long)ND * 32, TB), TB, 0, stream>>>(x_d, wvec, ad, ND, F_DIM);
    run_edges(ei_td, as, ad, hs, acc, ND);

    combine<<<cdiv((long)ND * C_DIM, TB), TB, 0, stream>>>(acc, b_dd, b_td, out_d, ND);
}